// SLAPS_17308718202950
// MI455X (gfx1250) — compile-verified
//
#include <hip/hip_runtime.h>
#include <hip/hip_bf16.h>
#include <math.h>

// ---------------------------------------------------------------------------
// GCN (2x GCNConv + log_softmax) for MI455X (gfx1250, wave32).
// Memory-bound problem (~1.1 GB traffic @ 23.3 TB/s ~= 50us); GEMMs are tiny
// (1.8 GFLOP) so we use full-precision V_WMMA_F32_16X16X4_F32 for them and
// spend the design effort on coalesced edge gather/scatter with native
// global_atomic_add_f32.
// ---------------------------------------------------------------------------

typedef float v2f __attribute__((ext_vector_type(2)));
typedef float v8f __attribute__((ext_vector_type(8)));

#define NFEAT 256
#define HID   32
#define NCLS  16

static __device__ __forceinline__ void atomic_add_f32(float* p, float v) {
  __hip_atomic_fetch_add(p, v, __ATOMIC_RELAXED, __HIP_MEMORY_SCOPE_AGENT);
}

// ---- init: zero layer accumulators, deg = 1.0 (self-loop weight) ----------
__global__ void gcn_init(float* __restrict__ agg1, float* __restrict__ acc2,
                         float* __restrict__ deg, int n) {
  int i = blockIdx.x * blockDim.x + threadIdx.x;
  if (i < n * HID)  agg1[i] = 0.0f;
  if (i < n * NCLS) acc2[i] = 0.0f;
  if (i < n)        deg[i]  = 1.0f;   // self-loop contributes weight 1
}

// ---- degree accumulation at target nodes ----------------------------------
__global__ void gcn_degree(const int* __restrict__ col,
                           const float* __restrict__ w,
                           float* __restrict__ deg, int E) {
  int e = blockIdx.x * blockDim.x + threadIdx.x;
  if (e < E) atomic_add_f32(&deg[col[e]], w[e]);
}

// ---- dinv = rsqrt(deg) in place -------------------------------------------
__global__ void gcn_dinv(float* __restrict__ deg, int n) {
  int i = blockIdx.x * blockDim.x + threadIdx.x;
  if (i < n) {
    float d = deg[i];
    deg[i] = d > 0.0f ? rsqrtf(d) : 0.0f;
  }
}

// ---- per-edge norm = dinv[row] * w * dinv[col] ----------------------------
__global__ void gcn_norm(const int* __restrict__ row, const int* __restrict__ col,
                         const float* __restrict__ w, const float* __restrict__ dinv,
                         float* __restrict__ norm, int E) {
  int e = blockIdx.x * blockDim.x + threadIdx.x;
  if (e < E) norm[e] = dinv[row[e]] * w[e] * dinv[col[e]];
}

// ---- GEMM1: XW = X[N,256] @ W1[256,32] via V_WMMA_F32_16X16X4_F32 ---------
// One wave owns a 16-row x 32-col output tile (two 16x16 accumulators).
// fp32 A 16x4 layout: lanes 0-15 hold {K=k,k+1}, lanes 16-31 hold {K=k+2,k+3}.
// fp32 C/D layout: VGPR r -> row m0+r (lanes 0-15) / m0+r+8 (lanes 16-31).
__global__ void gcn_gemm1(const float* __restrict__ X, const float* __restrict__ W,
                          float* __restrict__ XW, int ntiles) {
  int wave = (int)((blockIdx.x * blockDim.x + threadIdx.x) >> 5);
  int lane = threadIdx.x & 31;
  if (wave >= ntiles) return;           // uniform over the wave: EXEC stays all-1s
  int half = lane >> 4;                 // 0: lanes 0-15, 1: lanes 16-31
  int l    = lane & 15;
  int m0   = wave * 16;

  v8f acc0 = {};                        // cols 0-15
  v8f acc1 = {};                        // cols 16-31
  const float* xrow = X + (size_t)(m0 + l) * NFEAT;

  for (int k = 0; k < NFEAT; k += 4) {
    v2f a = *(const v2f*)(xrow + k + 2 * half);   // A frag (K = k+2*half, +1)
    const float* wk = W + (size_t)(k + 2 * half) * HID + l;
    v2f blo, bhi;
    blo.x = wk[0];        // (K = k+2*half,   col l)
    blo.y = wk[HID];      // (K = k+2*half+1, col l)
    bhi.x = wk[16];       // cols 16-31
    bhi.y = wk[HID + 16];
    acc0 = __builtin_amdgcn_wmma_f32_16x16x4_f32(false, a, false, blo,
                                                 (short)0, acc0, false, false);
    acc1 = __builtin_amdgcn_wmma_f32_16x16x4_f32(false, a, false, bhi,
                                                 (short)0, acc1, false, false);
  }

  float* out = XW + (size_t)(m0 + half * 8) * HID + l;
#pragma unroll
  for (int r = 0; r < 8; ++r) {
    out[(size_t)r * HID]      = acc0[r];
    out[(size_t)r * HID + 16] = acc1[r];
  }
}

// ---- layer-1 aggregation: one wave per edge, one lane per feature ---------
// Gather xw[row]*norm (coalesced 128B), atomic scatter-add into agg[col].
__global__ void gcn_agg32(const int* __restrict__ row, const int* __restrict__ col,
                          const float* __restrict__ norm,
                          const float* __restrict__ xw,
                          float* __restrict__ agg, int E) {
  int wave = (int)((blockIdx.x * blockDim.x + threadIdx.x) >> 5);
  int lane = threadIdx.x & 31;
  if (wave >= E) return;
  int r = row[wave];
  int c = col[wave];
  float nv = norm[wave];
  float v = xw[(size_t)r * HID + lane] * nv;
  atomic_add_f32(&agg[(size_t)c * HID + lane], v);
}

// ---- h = relu(agg + xw*dinv^2 + b1)  (fused self-loop term) ---------------
__global__ void gcn_relu(const float* __restrict__ agg, const float* __restrict__ xw,
                         const float* __restrict__ dinv, const float* __restrict__ b1,
                         float* __restrict__ h, int n) {
  int i = blockIdx.x * blockDim.x + threadIdx.x;
  if (i >= n * HID) return;
  int node = i >> 5;
  int f    = i & (HID - 1);
  float d = dinv[node];
  float v = agg[i] + xw[i] * d * d + b1[f];
  h[i] = v > 0.0f ? v : 0.0f;
}

// ---- GEMM2: HW = H[N,32] @ W2[32,16] via V_WMMA_F32_16X16X4_F32 -----------
__global__ void gcn_gemm2(const float* __restrict__ H, const float* __restrict__ W,
                          float* __restrict__ HW, int ntiles) {
  int wave = (int)((blockIdx.x * blockDim.x + threadIdx.x) >> 5);
  int lane = threadIdx.x & 31;
  if (wave >= ntiles) return;
  int half = lane >> 4;
  int l    = lane & 15;
  int m0   = wave * 16;

  v8f acc = {};
  const float* hrow = H + (size_t)(m0 + l) * HID;

  for (int k = 0; k < HID; k += 4) {
    v2f a = *(const v2f*)(hrow + k + 2 * half);
    const float* wk = W + (size_t)(k + 2 * half) * NCLS + l;
    v2f b;
    b.x = wk[0];
    b.y = wk[NCLS];
    acc = __builtin_amdgcn_wmma_f32_16x16x4_f32(false, a, false, b,
                                                (short)0, acc, false, false);
  }

  float* out = HW + (size_t)(m0 + half * 8) * NCLS + l;
#pragma unroll
  for (int r = 0; r < 8; ++r) out[(size_t)r * NCLS] = acc[r];
}

// ---- layer-2 aggregation: 16 lanes per edge -------------------------------
__global__ void gcn_agg16(const int* __restrict__ row, const int* __restrict__ col,
                          const float* __restrict__ norm,
                          const float* __restrict__ hw,
                          float* __restrict__ acc, int E) {
  int t = blockIdx.x * blockDim.x + threadIdx.x;
  int e = t >> 4;
  int f = t & 15;
  if (e >= E) return;
  int r = row[e];
  int c = col[e];
  float v = hw[(size_t)r * NCLS + f] * norm[e];
  atomic_add_f32(&acc[(size_t)c * NCLS + f], v);
}

// ---- logits = acc + hw*dinv^2 + b2 ; out = log_softmax(logits) ------------
__global__ void gcn_logsoftmax(const float* __restrict__ acc, const float* __restrict__ hw,
                               const float* __restrict__ dinv, const float* __restrict__ b2,
                               float* __restrict__ out, int n) {
  int i = blockIdx.x * blockDim.x + threadIdx.x;
  if (i >= n) return;
  float d  = dinv[i];
  float d2 = d * d;
  float v[NCLS];
  float m = -INFINITY;
#pragma unroll
  for (int f = 0; f < NCLS; ++f) {
    v[f] = acc[(size_t)i * NCLS + f] + hw[(size_t)i * NCLS + f] * d2 + b2[f];
    m = fmaxf(m, v[f]);
  }
  float s = 0.0f;
#pragma unroll
  for (int f = 0; f < NCLS; ++f) s += __expf(v[f] - m);
  float ls = m + __logf(s);
#pragma unroll
  for (int f = 0; f < NCLS; ++f) out[(size_t)i * NCLS + f] = v[f] - ls;
}

// ---------------------------------------------------------------------------
extern "C" void kernel_launch(void* const* d_in, const int* in_sizes, int n_in,
                              void* d_out, int out_size, void* d_ws, size_t ws_size,
                              hipStream_t stream) {
  // Inputs: x[N,256], edge_index[2,E] (int32: JAX x64 disabled), edge_weight[E],
  //         W1[256,32], b1[32], W2[32,16], b2[16]
  const float* x    = (const float*)d_in[0];
  const int*   eidx = (const int*)d_in[1];
  const float* ew   = (const float*)d_in[2];
  const float* W1   = (const float*)d_in[3];
  const float* b1   = (const float*)d_in[4];
  const float* W2   = (const float*)d_in[5];
  const float* b2   = (const float*)d_in[6];
  float* out = (float*)d_out;

  const int N = in_sizes[0] / NFEAT;       // 100000
  const int E = in_sizes[2];               // 3200000
  const int* erow = eidx;                  // edge_index[0]
  const int* ecol = eidx + E;              // edge_index[1]

  // Workspace layout (floats), ~64.5 MB total.
  float* ws = (float*)d_ws;
  size_t off = 0;
  auto take = [&](size_t cnt) { float* p = ws + off; off += (cnt + 63) & ~(size_t)63; return p; };
  float* deg  = take((size_t)N);           // degree -> dinv (in place)
  float* norm = take((size_t)E);
  float* xw   = take((size_t)N * HID);
  float* agg1 = take((size_t)N * HID);
  float* h    = take((size_t)N * HID);
  float* hw   = take((size_t)N * NCLS);
  float* acc2 = take((size_t)N * NCLS);
  (void)ws_size;

  const int B = 256;
  const int ntiles = (N + 15) / 16;                  // 6250 (N % 16 == 0)
  const int gemm_blocks = (ntiles + (B / 32) - 1) / (B / 32);

  gcn_init<<<(N * HID + B - 1) / B, B, 0, stream>>>(agg1, acc2, deg, N);
  gcn_degree<<<(E + B - 1) / B, B, 0, stream>>>(ecol, ew, deg, E);
  gcn_dinv<<<(N + B - 1) / B, B, 0, stream>>>(deg, N);
  gcn_norm<<<(E + B - 1) / B, B, 0, stream>>>(erow, ecol, ew, deg, norm, E);

  gcn_gemm1<<<gemm_blocks, B, 0, stream>>>(x, W1, xw, ntiles);
  gcn_agg32<<<(int)(((size_t)E * 32 + B - 1) / B), B, 0, stream>>>(erow, ecol, norm, xw, agg1, E);
  gcn_relu<<<(N * HID + B - 1) / B, B, 0, stream>>>(agg1, xw, deg, b1, h, N);

  gcn_gemm2<<<gemm_blocks, B, 0, stream>>>(h, W2, hw, ntiles);
  gcn_agg16<<<(int)(((size_t)E * 16 + B - 1) / B), B, 0, stream>>>(erow, ecol, norm, hw, acc2, E);
  gcn_logsoftmax<<<(N + B - 1) / B, B, 0, stream>>>(acc2, hw, deg, b2, out, N);
}